// MP_14001593385535
// MI455X (gfx1250) — compile-verified
//
#include <hip/hip_runtime.h>

#define B_   8
#define N_   4096
#define KNN_ 16
#define H_   4
#define DH_  16
#define D_   64
#define L_   4
#define NT_  (B_ * N_)

typedef __attribute__((ext_vector_type(2))) float v2f;
typedef __attribute__((ext_vector_type(8))) float v8f;

// ---------------------------------------------------------------------------
// kNN: per-graph top-16 nearest neighbors (excluding self), LDS-tiled.
// One thread per target node; sorted insertion keeps smallest-16 (stable ties
// by lower source index, matching jax.lax.top_k on -d2). Prefetch next tile.
// ---------------------------------------------------------------------------
__global__ __launch_bounds__(256) void knn_kernel(const float* __restrict__ c,
                                                  int* __restrict__ idx) {
  const int b = blockIdx.y;
  const int t = blockIdx.x * 256 + threadIdx.x;
  const float* cb = c + (size_t)b * N_ * 3;
  const float tx = cb[t * 3 + 0], ty = cb[t * 3 + 1], tz = cb[t * 3 + 2];
  const float tsq = tx * tx + ty * ty + tz * tz;

  __shared__ float4 tile[256];

  float dist[KNN_];
  int   ind [KNN_];
#pragma unroll
  for (int i = 0; i < KNN_; ++i) { dist[i] = 3.0e38f; ind[i] = 0; }

  for (int s0 = 0; s0 < N_; s0 += 256) {
    __syncthreads();
    {
      const int s = s0 + threadIdx.x;
      const float sx = cb[s * 3 + 0], sy = cb[s * 3 + 1], sz = cb[s * 3 + 2];
      tile[threadIdx.x] = make_float4(sx, sy, sz, sx * sx + sy * sy + sz * sz);
    }
    // Prefetch the next source tile while we grind on this one.
    if (s0 + 256 < N_)
      __builtin_prefetch(cb + (size_t)(s0 + 256 + threadIdx.x) * 3, 0, 0);
    __syncthreads();
    for (int j = 0; j < 256; ++j) {
      const int s = s0 + j;
      const float4 p = tile[j];
      const float d2 = tsq + p.w - 2.0f * (tx * p.x + ty * p.y + tz * p.z);
      if ((s != t) && (d2 < dist[KNN_ - 1])) {
        dist[KNN_ - 1] = d2; ind[KNN_ - 1] = s;
#pragma unroll
        for (int i = KNN_ - 1; i > 0; --i) {
          if (dist[i] < dist[i - 1]) {
            float td = dist[i]; dist[i] = dist[i - 1]; dist[i - 1] = td;
            int   ti = ind[i];  ind[i]  = ind[i - 1];  ind[i - 1]  = ti;
          }
        }
      }
    }
  }
#pragma unroll
  for (int i = 0; i < KNN_; ++i)
    idx[((size_t)b * N_ + t) * KNN_ + i] = ind[i];
}

// ---------------------------------------------------------------------------
// GEMM Y[M,DOUT] = X[M,DIN] @ W[DIN,DOUT] + bias, optional ReLU.
// Exact f32 via V_WMMA_F32_16X16X4_F32. One wave computes a 64x16 output
// slab = 4 independent 16x16 accumulator chains (hides WMMA RAW latency,
// amortizes the B-fragment load 4x).
// A 16x4 frag: lane L -> row M=L%16, K-pair 2*(L/16)+{0,1} per 4-chunk.
// B 4x16 frag: lane L -> col N=L%16, same K-pair striping.
// C/D: VGPR r -> M = r + 8*(L/16), N = L%16 within the 16x16 subtile.
// ---------------------------------------------------------------------------
template <int DIN, int DOUT, int RELU>
__global__ __launch_bounds__(128) void gemm_wmma(const float* __restrict__ X,
                                                 const float* __restrict__ W,
                                                 const float* __restrict__ bias,
                                                 float* __restrict__ Y) {
  const int lane = threadIdx.x & 31;
  const int wave = threadIdx.x >> 5;
  const int tile = blockIdx.x * 4 + wave;
  constexpr int TN = DOUT / 16;
  const int tM = tile / TN;           // 64-row slab index
  const int tN = tile - tM * TN;      // 16-col slab index
  const int l16 = lane & 15;
  const int hi  = lane >> 4;          // 0: K {0,1}, 1: K {2,3} within 4-chunk

  const int n = tN * 16 + l16;
  const float* xr0 = X + (size_t)(tM * 64 +  0 + l16) * DIN + 2 * hi;
  const float* xr1 = X + (size_t)(tM * 64 + 16 + l16) * DIN + 2 * hi;
  const float* xr2 = X + (size_t)(tM * 64 + 32 + l16) * DIN + 2 * hi;
  const float* xr3 = X + (size_t)(tM * 64 + 48 + l16) * DIN + 2 * hi;
  const float* wcol = W + (size_t)(2 * hi) * DOUT + n;

  v8f acc0 = {}, acc1 = {}, acc2 = {}, acc3 = {};
#pragma unroll
  for (int k0 = 0; k0 < DIN; k0 += 4) {
    v2f bf;
    bf.x = wcol[(size_t)k0 * DOUT];
    bf.y = wcol[(size_t)(k0 + 1) * DOUT];
    v2f a0, a1, a2, a3;
    a0.x = xr0[k0]; a0.y = xr0[k0 + 1];
    a1.x = xr1[k0]; a1.y = xr1[k0 + 1];
    a2.x = xr2[k0]; a2.y = xr2[k0 + 1];
    a3.x = xr3[k0]; a3.y = xr3[k0 + 1];
    acc0 = __builtin_amdgcn_wmma_f32_16x16x4_f32(false, a0, false, bf, (short)0, acc0, false, false);
    acc1 = __builtin_amdgcn_wmma_f32_16x16x4_f32(false, a1, false, bf, (short)0, acc1, false, false);
    acc2 = __builtin_amdgcn_wmma_f32_16x16x4_f32(false, a2, false, bf, (short)0, acc2, false, false);
    acc3 = __builtin_amdgcn_wmma_f32_16x16x4_f32(false, a3, false, bf, (short)0, acc3, false, false);
  }

  const float bn = bias[n];
  const int rowBase = tM * 64 + 8 * hi;
#pragma unroll
  for (int r = 0; r < 8; ++r) {
    float v0 = acc0[r] + bn;
    float v1 = acc1[r] + bn;
    float v2 = acc2[r] + bn;
    float v3 = acc3[r] + bn;
    if (RELU) {
      v0 = fmaxf(v0, 0.0f); v1 = fmaxf(v1, 0.0f);
      v2 = fmaxf(v2, 0.0f); v3 = fmaxf(v3, 0.0f);
    }
    Y[(size_t)(rowBase + r +  0) * DOUT + n] = v0;
    Y[(size_t)(rowBase + r + 16) * DOUT + n] = v1;
    Y[(size_t)(rowBase + r + 32) * DOUT + n] = v2;
    Y[(size_t)(rowBase + r + 48) * DOUT + n] = v3;
  }
}

// ---------------------------------------------------------------------------
// Edge attention: one thread per (node, head). 16 neighbors, DH=16.
// score = clamp(q.k / 4, -5, 5); e = exp; out = sum(e*v) / (sum(e)+1e-6)
// ---------------------------------------------------------------------------
__global__ __launch_bounds__(256) void attn_kernel(const float* __restrict__ Q,
                                                   const float* __restrict__ K,
                                                   const float* __restrict__ V,
                                                   const int* __restrict__ idx,
                                                   float* __restrict__ O) {
  const int t = blockIdx.x * 256 + threadIdx.x;
  const int node = t >> 2;
  const int head = t & 3;
  const int b = node >> 12;  // N_ = 4096

  const float* q = Q + (size_t)node * D_ + head * DH_;
  float qr[DH_];
#pragma unroll
  for (int d = 0; d < DH_; ++d) qr[d] = q[d];

  float acc[DH_];
#pragma unroll
  for (int d = 0; d < DH_; ++d) acc[d] = 0.0f;
  float denom = 0.0f;

  const int* ip = idx + (size_t)node * KNN_;
  for (int j = 0; j < KNN_; ++j) {
    const int s = (b << 12) + ip[j];
    const float* kp = K + (size_t)s * D_ + head * DH_;
    float dot = 0.0f;
#pragma unroll
    for (int d = 0; d < DH_; ++d) dot += qr[d] * kp[d];
    float sc = dot * 0.25f;  // 1/sqrt(DH)
    sc = fminf(fmaxf(sc, -5.0f), 5.0f);
    const float e = expf(sc);
    denom += e;
    const float* vp = V + (size_t)s * D_ + head * DH_;
#pragma unroll
    for (int d = 0; d < DH_; ++d) acc[d] += e * vp[d];
  }

  const float inv = 1.0f / (denom + 1e-6f);
  float* op = O + (size_t)node * D_ + head * DH_;
#pragma unroll
  for (int d = 0; d < DH_; ++d) op[d] = acc[d] * inv;
}

// ---------------------------------------------------------------------------
// Residual add + LayerNorm over D=64, one thread per node.
// ---------------------------------------------------------------------------
__global__ __launch_bounds__(256) void ln_kernel(const float* __restrict__ resid,
                                                 const float* __restrict__ x,
                                                 const float* __restrict__ g,
                                                 const float* __restrict__ be,
                                                 float* __restrict__ out) {
  const int nIdx = blockIdx.x * 256 + threadIdx.x;
  const float* rp = resid + (size_t)nIdx * D_;
  const float* xp = x + (size_t)nIdx * D_;

  float v[D_];
  float m = 0.0f;
#pragma unroll
  for (int d = 0; d < D_; ++d) { v[d] = rp[d] + xp[d]; m += v[d]; }
  m *= (1.0f / D_);
  float var = 0.0f;
#pragma unroll
  for (int d = 0; d < D_; ++d) { float c0 = v[d] - m; var += c0 * c0; }
  var *= (1.0f / D_);
  const float r = rsqrtf(var + 1e-5f);

  float* op = out + (size_t)nIdx * D_;
#pragma unroll
  for (int d = 0; d < D_; ++d) op[d] = (v[d] - m) * r * g[d] + be[d];
}

// ---------------------------------------------------------------------------
extern "C" void kernel_launch(void* const* d_in, const int* in_sizes, int n_in,
                              void* d_out, int out_size, void* d_ws, size_t ws_size,
                              hipStream_t stream) {
  (void)in_sizes; (void)n_in; (void)out_size; (void)ws_size;
  const float* h    = (const float*)d_in[0];
  const float* c    = (const float*)d_in[1];
  const float* Wemb = (const float*)d_in[2];
  const float* bemb = (const float*)d_in[3];
  const float* Wq   = (const float*)d_in[4];
  const float* bq   = (const float*)d_in[5];
  const float* Wk   = (const float*)d_in[6];
  const float* bk   = (const float*)d_in[7];
  const float* Wv   = (const float*)d_in[8];
  const float* bv   = (const float*)d_in[9];
  const float* Wo   = (const float*)d_in[10];
  const float* bo   = (const float*)d_in[11];
  const float* g1   = (const float*)d_in[12];
  const float* be1  = (const float*)d_in[13];
  const float* W1   = (const float*)d_in[14];
  const float* bf1  = (const float*)d_in[15];
  const float* W2   = (const float*)d_in[16];
  const float* bf2  = (const float*)d_in[17];
  const float* g2   = (const float*)d_in[18];
  const float* be2  = (const float*)d_in[19];
  float* out = (float*)d_out;

  // Workspace layout (aliased): idx | hbuf | q | k | v | attn
  //   ffh (NT x 128) aliases [v, attn]; proj aliases q; ff2 aliases k.
  char* ws = (char*)d_ws;
  int*   idx  = (int*)ws;
  float* hbuf = (float*)(ws + (size_t)NT_ * KNN_ * sizeof(int));
  float* qb   = hbuf + (size_t)NT_ * D_;
  float* kb   = qb   + (size_t)NT_ * D_;
  float* vb   = kb   + (size_t)NT_ * D_;
  float* ab   = vb   + (size_t)NT_ * D_;
  float* ffh  = vb;   // spans NT_*128 floats across vb..ab
  float* proj = qb;
  float* ff2  = kb;

  knn_kernel<<<dim3(N_ / 256, B_), 256, 0, stream>>>(c, idx);

  // Tile-group count: (NT/64) * (DOUT/16) waves, 4 waves per block.
  constexpr int G64  = (NT_ / 64) * (64 / 16) / 4;   // DOUT=64  -> 512 blocks
  constexpr int G128 = (NT_ / 64) * (128 / 16) / 4;  // DOUT=128 -> 1024 blocks

  // Embedding: [NT,16] @ [16,64] + b
  gemm_wmma<16, 64, 0><<<G64, 128, 0, stream>>>(h, Wemb, bemb, hbuf);

  for (int l = 0; l < L_; ++l) {
    gemm_wmma<64, 64, 0><<<G64, 128, 0, stream>>>(hbuf, Wq + l * 64 * 64, bq + l * 64, qb);
    gemm_wmma<64, 64, 0><<<G64, 128, 0, stream>>>(hbuf, Wk + l * 64 * 64, bk + l * 64, kb);
    gemm_wmma<64, 64, 0><<<G64, 128, 0, stream>>>(hbuf, Wv + l * 64 * 64, bv + l * 64, vb);

    attn_kernel<<<NT_ * H_ / 256, 256, 0, stream>>>(qb, kb, vb, idx, ab);

    gemm_wmma<64, 64, 0><<<G64, 128, 0, stream>>>(ab, Wo + l * 64 * 64, bo + l * 64, proj);
    ln_kernel<<<NT_ / 256, 256, 0, stream>>>(hbuf, proj, g1 + l * 64, be1 + l * 64, hbuf);

    gemm_wmma<64, 128, 1><<<G128, 128, 0, stream>>>(hbuf, W1 + l * 64 * 128, bf1 + l * 128, ffh);
    gemm_wmma<128, 64, 0><<<G64, 128, 0, stream>>>(ffh, W2 + l * 128 * 64, bf2 + l * 64, ff2);
    ln_kernel<<<NT_ / 256, 256, 0, stream>>>(hbuf, ff2, g2 + l * 64, be2 + l * 64,
                                             (l == L_ - 1) ? out : hbuf);
  }
}